// PNN1_35716948034260
// MI455X (gfx1250) — compile-verified
//
#include <hip/hip_runtime.h>

typedef float v2f __attribute__((ext_vector_type(2)));
typedef float v4f __attribute__((ext_vector_type(4)));
typedef float v8f __attribute__((ext_vector_type(8)));

#define NF   39
#define NV   200000
#define EPAD 12
#define KP   (NF * EPAD)   /* 468 padded K */
#define NKC  (KP / 4)      /* 117 k-chunks */
#define ST   34            /* LDS row stride for activation tile */
#define NW   8             /* waves per block */
#define ROWS (NW * 16)     /* 128 batch rows per block */

#define WPAD_ELEMS (NKC * 32 * 8)      /* 29952 floats: [kc][n][g][mat][j] */
#define LLIN_ELEMS (2 * 8 * 32 * 4)    /* 2048 floats:  [layer][kc][n][g][j] */

static __device__ __forceinline__ v8f wmma4(v2f a, v2f b, v8f c) {
  // D(16x16,f32) = A(16x4,f32) * B(4x16,f32) + C
  return __builtin_amdgcn_wmma_f32_16x16x4_f32(
      /*neg_a=*/false, a, /*neg_b=*/false, b,
      /*c_mod=*/(short)0, c, /*reuse_a=*/false, /*reuse_b=*/false);
}

static __device__ __forceinline__ v2f lo2(v4f q) {
  return __builtin_shufflevector(q, q, 0, 1);
}
static __device__ __forceinline__ v2f hi2(v4f q) {
  return __builtin_shufflevector(q, q, 2, 3);
}

// Repack W1/W2 (D0=32, F=39, E=10) into per-lane WMMA B-operand order:
// Wpad[((kc*32 + n)*2 + g)*4 + mat*2 + j] = W_mat[n][k], k = 4*kc + 2*g + j
__global__ void pnn_prep(const float* __restrict__ W1, const float* __restrict__ W2,
                         float* __restrict__ Wpad) {
  int i = blockIdx.x * blockDim.x + threadIdx.x;
  if (i >= WPAD_ELEMS) return;
  int j   = i & 1;
  int mat = (i >> 1) & 1;
  int g   = (i >> 2) & 1;
  int n   = (i >> 3) & 31;
  int kc  = i >> 8;
  int k = 4 * kc + 2 * g + j;
  int f = k / EPAD, e = k % EPAD;
  const float* W = mat ? W2 : W1;
  Wpad[i] = (e < 10) ? W[n * 390 + f * 10 + e] : 0.0f;
}

__launch_bounds__(256)
__global__ void pnn_main(const long long* __restrict__ Xi, const float* __restrict__ Xv,
                         const float* __restrict__ emb, const float* __restrict__ Wpad,
                         const float* __restrict__ l1w, const float* __restrict__ l1b,
                         const float* __restrict__ l2w, const float* __restrict__ l2b,
                         const float* __restrict__ lastw, const float* __restrict__ lastb,
                         float* __restrict__ out) {
  __shared__ __attribute__((aligned(16))) float LW[WPAD_ELEMS];   // 119.8 KB
  __shared__ __attribute__((aligned(16))) float LLIN[LLIN_ELEMS]; // 8 KB
  __shared__ float LB[64];
  __shared__ float LLAST[32];
  __shared__ float LX[NW * 16 * ST];                              // 17.4 KB

  const int tid = threadIdx.x;

  // ---- Phase A: LDS fill ----
  {
    const v4f* s4 = (const v4f*)Wpad;
    v4f* d4 = (v4f*)LW;
    for (int i = tid; i < WPAD_ELEMS / 4; i += 256) d4[i] = s4[i];
  }
  for (int i = tid; i < LLIN_ELEMS; i += 256) {
    int j     = i & 1;
    int g     = (i >> 1) & 1;
    int n     = (i >> 2) & 31;
    int kc    = (i >> 7) & 7;
    int layer = i >> 10;
    int k = 4 * kc + 2 * g + j;
    const float* w = layer ? l2w : l1w;
    LLIN[i] = w[n * 32 + k];  // B[k][n] = w[n][k]  (y = x @ w^T)
  }
  if (tid < 32) {
    LB[tid]      = l1b[tid];
    LB[32 + tid] = l2b[tid];
    LLAST[tid]   = lastw[tid];
  }
  __syncthreads();

  // ---- Phase B: per-wave 16-row tile ----
  const int lane = tid & 31;
  const int w    = tid >> 5;
  const int r    = lane & 15;   // tile row (A/M); also N-within-half for B/C
  const int g    = lane >> 4;   // half-wave: owns K pair {2g, 2g+1} of each chunk
  const int b0   = blockIdx.x * ROWS + w * 16;
  const int row  = b0 + r;
  const int base = row * NF;

  v8f a1h0 = {}, a1h1 = {}, a2h0 = {}, a2h1 = {};

  const v4f* Bp    = (const v4f*)LW;      // [(kc*32 + n)*2 + g] -> {W1 pair | W2 pair}
  const float2* E2 = (const float2*)emb;  // 32-bit indexable: whole table < 2^31 bytes
  const int* Xi32  = (const int*)Xi;      // low dword of little-endian int64

  // ---- software pipeline: idx/xv two features ahead, gathers one ahead ----
  int   idxN = Xi32[(base + 0) * 2];      // index for f (next to gather)
  float xvN  = Xv[base + 0];
  int   f1n  = 0;                         // feature number matching idxN

  // issue gathers for f = 0
  int o0 = idxN * 5;
  float2 tA0 = E2[o0 + g];                // pair e{2g, 2g+1}
  float2 tA1 = E2[o0 + 2 + g];            // pair e{4+2g, 5+2g}
  float2 tA2 = E2[o0 + 4];                // pair e{8,9} (pad pair zeroed via scale)
  float xvA  = xvN;

  // idx/xv for f = 1
  idxN = Xi32[(base + 1) * 2];
  xvN  = Xv[base + 1];

#pragma clang loop unroll(disable)
  for (int f = 0; f < NF; ++f) {
    // issue gathers for feature f+1 (index already resident)
    int f1 = (f + 1 < NF) ? f + 1 : NF - 1;
    int oB = f1 * (NV * 5) + idxN * 5;
    float2 tB0 = E2[oB + g];
    float2 tB1 = E2[oB + 2 + g];
    float2 tB2 = E2[oB + 4];
    float xvB  = xvN;
    // issue idx/xv load for feature f+2
    int f2 = (f + 2 < NF) ? f + 2 : NF - 1;
    idxN = Xi32[(base + f2) * 2];
    xvN  = Xv[base + f2];

    // compute feature f
    float scC = g ? 0.0f : xvA;
    v2f aA0, aA1, aA2;
    aA0.x = tA0.x * xvA;  aA0.y = tA0.y * xvA;
    aA1.x = tA1.x * xvA;  aA1.y = tA1.y * xvA;
    aA2.x = tA2.x * scC;  aA2.y = tA2.y * scC;

    int kc0 = 3 * f;
    {
      v4f q0 = Bp[((kc0 + 0) * 32 + r) * 2 + g];
      v4f q1 = Bp[((kc0 + 0) * 32 + 16 + r) * 2 + g];
      a1h0 = wmma4(aA0, lo2(q0), a1h0);
      a1h1 = wmma4(aA0, lo2(q1), a1h1);
      a2h0 = wmma4(aA0, hi2(q0), a2h0);
      a2h1 = wmma4(aA0, hi2(q1), a2h1);
    }
    {
      v4f q0 = Bp[((kc0 + 1) * 32 + r) * 2 + g];
      v4f q1 = Bp[((kc0 + 1) * 32 + 16 + r) * 2 + g];
      a1h0 = wmma4(aA1, lo2(q0), a1h0);
      a1h1 = wmma4(aA1, lo2(q1), a1h1);
      a2h0 = wmma4(aA1, hi2(q0), a2h0);
      a2h1 = wmma4(aA1, hi2(q1), a2h1);
    }
    {
      v4f q0 = Bp[((kc0 + 2) * 32 + r) * 2 + g];
      v4f q1 = Bp[((kc0 + 2) * 32 + 16 + r) * 2 + g];
      a1h0 = wmma4(aA2, lo2(q0), a1h0);
      a1h1 = wmma4(aA2, lo2(q1), a1h1);
      a2h0 = wmma4(aA2, hi2(q0), a2h0);
      a2h1 = wmma4(aA2, hi2(q1), a2h1);
    }

    tA0 = tB0; tA1 = tB1; tA2 = tB2; xvA = xvB;
  }

  // x = first + s*s (C-layout accumulators)
  v8f x0 = a1h0 + a2h0 * a2h0;
  v8f x1 = a1h1 + a2h1 * a2h1;

  // C-layout -> LDS tile: element e at (row = e + 8g, col = r + 16h)
  float* X = LX + w * 16 * ST;
#pragma unroll
  for (int e = 0; e < 8; ++e) {
    X[(e + 8 * g) * ST + r]      = x0[e];
    X[(e + 8 * g) * ST + 16 + r] = x1[e];
  }
  __syncthreads();

  // ---- MLP: two 32x32 WMMA layers, bias + ReLU ----
  const v2f* Lp = (const v2f*)LLIN;  // [((layer*8+kc)*32 + n)*2 + g]
  v8f y0, y1;
#pragma unroll
  for (int layer = 0; layer < 2; ++layer) {
    float bv0 = LB[layer * 32 + r];
    float bv1 = LB[layer * 32 + 16 + r];
#pragma unroll
    for (int e = 0; e < 8; ++e) { y0[e] = bv0; y1[e] = bv1; }
#pragma unroll
    for (int kc = 0; kc < 8; ++kc) {
      int kb = 4 * kc + 2 * g;
      v2f a;
      a.x = X[r * ST + kb];
      a.y = X[r * ST + kb + 1];
      v2f b0v = Lp[((layer * 8 + kc) * 32 + r) * 2 + g];
      v2f b1v = Lp[((layer * 8 + kc) * 32 + 16 + r) * 2 + g];
      y0 = wmma4(a, b0v, y0);
      y1 = wmma4(a, b1v, y1);
    }
#pragma unroll
    for (int e = 0; e < 8; ++e) {
      y0[e] = fmaxf(y0[e], 0.0f);
      y1[e] = fmaxf(y1[e], 0.0f);
    }
    __syncthreads();
#pragma unroll
    for (int e = 0; e < 8; ++e) {
      X[(e + 8 * g) * ST + r]      = y0[e];
      X[(e + 8 * g) * ST + 16 + r] = y1[e];
    }
    __syncthreads();
  }

  // ---- last layer (32 -> 1) + row sum: dot with last_w, butterfly over 16 lanes ----
  float lw0 = LLAST[r];
  float lw1 = LLAST[16 + r];
  float p[8];
#pragma unroll
  for (int e = 0; e < 8; ++e) p[e] = y0[e] * lw0 + y1[e] * lw1;
#pragma unroll
  for (int m = 1; m <= 8; m <<= 1) {
#pragma unroll
    for (int e = 0; e < 8; ++e) p[e] += __shfl_xor(p[e], m, 32);
  }
  if (r == 0) {
    float lb = lastb[0];
#pragma unroll
    for (int e = 0; e < 8; ++e) out[b0 + e + 8 * g] = p[e] + lb;
  }
}

extern "C" void kernel_launch(void* const* d_in, const int* in_sizes, int n_in,
                              void* d_out, int out_size, void* d_ws, size_t ws_size,
                              hipStream_t stream) {
  const long long* Xi   = (const long long*)d_in[0];   // int64 (B,F,1)
  const float* Xv       = (const float*)d_in[1];
  const float* emb      = (const float*)d_in[2];
  const float* W1       = (const float*)d_in[3];
  const float* W2       = (const float*)d_in[4];
  const float* l1w      = (const float*)d_in[5];
  const float* l1b      = (const float*)d_in[6];
  const float* l2w      = (const float*)d_in[7];
  const float* l2b      = (const float*)d_in[8];
  const float* lastw    = (const float*)d_in[9];
  const float* lastb    = (const float*)d_in[10];
  float* out            = (float*)d_out;
  float* Wpad           = (float*)d_ws;                // needs 29952*4 = 117 KB

  pnn_prep<<<(WPAD_ELEMS + 255) / 256, 256, 0, stream>>>(W1, W2, Wpad);
  pnn_main<<<16384 / ROWS, 256, 0, stream>>>(Xi, Xv, emb, Wpad,
                                             l1w, l1b, l2w, l2b, lastw, lastb, out);
}